// GCN_PYG_26912265077117
// MI455X (gfx1250) — compile-verified
//
#include <hip/hip_runtime.h>

typedef __attribute__((ext_vector_type(2))) float v2f;
typedef __attribute__((ext_vector_type(8))) float v8f;

// ---------------------------------------------------------------------------
// WMMA GEMM: C[N x 64] = op(A)[N x K] @ B[K x 64] (+ bias)
// op(A) = relu(A) if RELU_A. One wave computes one 16x16 tile via
// V_WMMA_F32_16X16X4_F32; a 128-thread block (4 waves) covers 16 rows x 64 cols.
// A-matrix lane layout (16x4 f32): lane L holds A[L%16][2*(L/16)+{0,1}].
// B-matrix (4x16): lane L holds B[2*(L/16)+{0,1}][L%16].
// C/D (16x16, 8 VGPRs): lane L, vgpr v -> row v + 8*(L/16), col L%16.
// ---------------------------------------------------------------------------
template <bool RELU_A>
__global__ __launch_bounds__(128) void gemm_wmma_kernel(
    const float* __restrict__ A, const float* __restrict__ B,
    const float* __restrict__ bias, float* __restrict__ C, int K) {
  const int lane = threadIdx.x & 31;
  const int wave = threadIdx.x >> 5;   // column tile 0..3
  const int row0 = blockIdx.x << 4;    // 16 rows per block
  const int col0 = wave << 4;
  const int r    = lane & 15;
  const int half = lane >> 4;          // 0 or 1

  const float* Arow = A + (size_t)(row0 + r) * K;
  v8f acc = {};

#pragma unroll 4
  for (int k = 0; k < K; k += 4) {
    const int ka = k + (half << 1);
    float ax = Arow[ka];
    float ay = Arow[ka + 1];
    if (RELU_A) {
      ax = fmaxf(ax, 0.0f);
      ay = fmaxf(ay, 0.0f);
    }
    v2f a, b;
    a.x = ax;
    a.y = ay;
    b.x = B[(size_t)ka * 64 + col0 + r];
    b.y = B[(size_t)(ka + 1) * 64 + col0 + r];
    // 8 args: (neg_a, A, neg_b, B, c_mod, C, reuse_a, reuse_b)
    acc = __builtin_amdgcn_wmma_f32_16x16x4_f32(false, a, false, b, (short)0,
                                                acc, false, false);
  }

  const float bv = bias ? bias[col0 + r] : 0.0f;
#pragma unroll
  for (int v = 0; v < 8; ++v) {
    C[(size_t)(row0 + v + (half << 3)) * 64 + col0 + r] = acc[v] + bv;
  }
}

// ---------------------------------------------------------------------------
// Degree / normalization helpers
// ---------------------------------------------------------------------------
__global__ void deg_init_kernel(float* __restrict__ deg, int n) {
  int i = blockIdx.x * blockDim.x + threadIdx.x;
  if (i < n) deg[i] = 1.0f;  // self-loop contributes 1 to every node's degree
}

__global__ void deg_accum_kernel(const long long* __restrict__ dst,
                                 float* __restrict__ deg, int e) {
  int i = blockIdx.x * blockDim.x + threadIdx.x;
  if (i < e) atomicAdd(&deg[(size_t)dst[i]], 1.0f);
}

__global__ void dinv_kernel(const float* __restrict__ deg,
                            float* __restrict__ dinv, int n) {
  int i = blockIdx.x * blockDim.x + threadIdx.x;
  if (i < n) {
    float d = deg[i];
    dinv[i] = (d > 0.0f) ? rsqrtf(d) : 0.0f;
  }
}

// ---------------------------------------------------------------------------
// Aggregation init: out[i][:] = bias + t[i][:] * dinv[i]^2  (self-loop fused)
// One thread per float4 chunk: 16 threads cover one node's 64 features.
// ---------------------------------------------------------------------------
__global__ void agg_init_kernel(const float* __restrict__ t,
                                const float* __restrict__ dinv,
                                const float* __restrict__ bias,
                                float* __restrict__ out, int n) {
  int i = blockIdx.x * blockDim.x + threadIdx.x;
  if (i >= n * 16) return;
  int node = i >> 4;
  int c = i & 15;
  float d = dinv[node];
  float s = d * d;
  float4 v = ((const float4*)(t + (size_t)node * 64))[c];
  float4 b4 = ((const float4*)bias)[c];
  float4 o;
  o.x = b4.x + v.x * s;
  o.y = b4.y + v.y * s;
  o.z = b4.z + v.z * s;
  o.w = b4.w + v.w * s;
  ((float4*)(out + (size_t)node * 64))[c] = o;
}

// ---------------------------------------------------------------------------
// Edge scatter: out[dst] += t[src] * dinv[src]*dinv[dst]
// 16 threads per edge, float4 gather (coalesced 256B per edge), fp32 atomics
// resolved in L2 (all tensors fit in the 192MB device L2).
// ---------------------------------------------------------------------------
__global__ void agg_edges_kernel(const long long* __restrict__ src,
                                 const long long* __restrict__ dst,
                                 const float* __restrict__ t,
                                 const float* __restrict__ dinv,
                                 float* __restrict__ out, int e) {
  int i = blockIdx.x * blockDim.x + threadIdx.x;
  if (i >= e * 16) return;
  int ed = i >> 4;
  int c = i & 15;
  long long s = src[ed];
  long long d = dst[ed];
  float nrm = dinv[s] * dinv[d];
  float4 v = ((const float4*)(t + (size_t)s * 64))[c];
  float* o = out + (size_t)d * 64 + (size_t)c * 4;
  atomicAdd(o + 0, v.x * nrm);
  atomicAdd(o + 1, v.y * nrm);
  atomicAdd(o + 2, v.z * nrm);
  atomicAdd(o + 3, v.w * nrm);
}

// ---------------------------------------------------------------------------
// Row-wise L2 normalize in place. One wave32 per row, 2 feats per lane,
// __shfl_xor butterfly reduction.
// ---------------------------------------------------------------------------
__global__ void l2norm_kernel(float* __restrict__ out, int n) {
  int row = blockIdx.x * (blockDim.x >> 5) + (threadIdx.x >> 5);
  int lane = threadIdx.x & 31;
  if (row >= n) return;
  float2* p = (float2*)(out + (size_t)row * 64);
  float2 v = p[lane];
  float ss = v.x * v.x + v.y * v.y;
#pragma unroll
  for (int off = 16; off > 0; off >>= 1) ss += __shfl_xor(ss, off, 32);
  float nrm = fmaxf(sqrtf(ss), 1e-12f);
  float inv = 1.0f / nrm;
  v.x *= inv;
  v.y *= inv;
  p[lane] = v;
}

// ---------------------------------------------------------------------------
extern "C" void kernel_launch(void* const* d_in, const int* in_sizes, int n_in,
                              void* d_out, int out_size, void* d_ws,
                              size_t ws_size, hipStream_t stream) {
  const float* x          = (const float*)d_in[0];      // [N, 128]
  const long long* eidx   = (const long long*)d_in[1];  // [2, E] int64
  const float* W_pre      = (const float*)d_in[2];      // [128, 64]
  const float* b_pre      = (const float*)d_in[3];      // [64]
  const float* W1         = (const float*)d_in[4];      // [64, 64]
  const float* b1         = (const float*)d_in[5];
  const float* Wh         = (const float*)d_in[6];
  const float* bh         = (const float*)d_in[7];
  const float* Wo         = (const float*)d_in[8];
  const float* bo         = (const float*)d_in[9];

  const int N = in_sizes[0] / 128;  // 100000 (multiple of 16)
  const int E = in_sizes[1] / 2;    // 1600000
  const long long* src = eidx;
  const long long* dst = eidx + E;

  float* bufA = (float*)d_ws;              // N*64
  float* bufB = bufA + (size_t)N * 64;     // N*64
  float* bufC = bufB + (size_t)N * 64;     // N*64
  float* deg  = bufC + (size_t)N * 64;     // N
  float* dinv = deg + N;                   // N
  float* outp = (float*)d_out;             // N*64

  const int TB = 256;
  const int gN    = (N + TB - 1) / TB;
  const int gE    = (E + TB - 1) / TB;
  const int gN16  = (N * 16 + TB - 1) / TB;
  const int gE16  = (int)(((long long)E * 16 + TB - 1) / TB);
  const int gGemm = N / 16;                 // exact: 100000 % 16 == 0
  const int gNorm = (N + 7) / 8;            // 8 rows (waves) per 256-thr block

  // --- degree & symmetric normalization ---
  deg_init_kernel<<<gN, TB, 0, stream>>>(deg, N);
  deg_accum_kernel<<<gE, TB, 0, stream>>>(dst, deg, E);
  dinv_kernel<<<gN, TB, 0, stream>>>(deg, dinv, N);

  // --- feature_pre: bufA = x @ W_pre + b_pre  (K=128) ---
  gemm_wmma_kernel<false><<<gGemm, 128, 0, stream>>>(x, W_pre, b_pre, bufA, 128);

  // --- conv1: bufB = bufA @ W1 ; bufC = b1 + scatter(bufB) ---
  gemm_wmma_kernel<false><<<gGemm, 128, 0, stream>>>(bufA, W1, nullptr, bufB, 64);
  agg_init_kernel<<<gN16, TB, 0, stream>>>(bufB, dinv, b1, bufC, N);
  agg_edges_kernel<<<gE16, TB, 0, stream>>>(src, dst, bufB, dinv, bufC, E);

  // --- conv2: bufA = relu(bufC) @ Wh ; bufB = bh + scatter(bufA) ---
  gemm_wmma_kernel<true><<<gGemm, 128, 0, stream>>>(bufC, Wh, nullptr, bufA, 64);
  agg_init_kernel<<<gN16, TB, 0, stream>>>(bufA, dinv, bh, bufB, N);
  agg_edges_kernel<<<gE16, TB, 0, stream>>>(src, dst, bufA, dinv, bufB, E);

  // --- conv3: bufC = relu(bufB) @ Wo ; d_out = bo + scatter(bufC) ---
  gemm_wmma_kernel<true><<<gGemm, 128, 0, stream>>>(bufB, Wo, nullptr, bufC, 64);
  agg_init_kernel<<<gN16, TB, 0, stream>>>(bufC, dinv, bo, outp, N);
  agg_edges_kernel<<<gE16, TB, 0, stream>>>(src, dst, bufC, dinv, outp, E);

  // --- row-wise L2 normalize in place ---
  l2norm_kernel<<<gNorm, TB, 0, stream>>>(outp, N);
}